// Decoder_C2F_88699664597871
// MI455X (gfx1250) — compile-verified
//
#include <hip/hip_runtime.h>
#include <math.h>

// ---------------- problem constants ----------------
#define NPTS   2048
#define BATCH  4
#define PTOT   (BATCH*NPTS)    // 8192 coarse/input points
#define PNUM   16384           // fine points per batch
#define PALL   (BATCH*PNUM)    // 65536 flattened points (== batchnorm reduce axis)
#define KPAD   40              // compact fc1 K (37 real channels, padded to mult of 4)
#define C1     512
#define BIGF   3.402823466e38f

// output layout (flat concat of the reference tuple)
#define OUT_COARSE 0
#define OUT_FINE   24576
#define OUT_DENSE  221184
#define OUT_NORM   229376

typedef float v2f __attribute__((ext_vector_type(2)));
typedef float v8f __attribute__((ext_vector_type(8)));

// ================= fp32 WMMA GEMM: C[M][N] = A[M][K] * B[K][N] (+add) ================
// one wave per 16(M) x 64(N) tile; K stepped by 4 (V_WMMA_F32_16X16X4_F32).
// addMode: 0 none, 1 += rowAdd[m], 2 += rowAdd[m*4 + (n>>14)]  (per-batch z-term)
__global__ __launch_bounds__(32)
void wmma_gemm_f32(const float* __restrict__ A, const float* __restrict__ B,
                   float* __restrict__ C, const float* __restrict__ rowAdd,
                   int M, int K, int Nn, int addMode)
{
  const int lane = threadIdx.x;        // 0..31
  const int half = lane >> 4;          // A/B K-pair select: lanes 0-15 -> K{0,1}, 16-31 -> K{2,3}
  const int l16  = lane & 15;
  const int m0 = blockIdx.y * 16;
  const int n0 = blockIdx.x * 64;

  v8f acc0 = {}; v8f acc1 = {}; v8f acc2 = {}; v8f acc3 = {};

  for (int k0 = 0; k0 < K; k0 += 4) {
    // A tile 16x4: lane holds row m = m0+l16, elements K = k0+2*half+{0,1}
    const float* arow = A + (size_t)(m0 + l16) * K + (k0 + 2*half);
    v2f a; a.x = arow[0]; a.y = arow[1];
    const float* brow0 = B + (size_t)(k0 + 2*half)     * Nn + n0 + l16;
    const float* brow1 = B + (size_t)(k0 + 2*half + 1) * Nn + n0 + l16;
    v2f b;
    b.x = brow0[0];  b.y = brow1[0];
    acc0 = __builtin_amdgcn_wmma_f32_16x16x4_f32(false, a, false, b, (short)0, acc0, false, false);
    b.x = brow0[16]; b.y = brow1[16];
    acc1 = __builtin_amdgcn_wmma_f32_16x16x4_f32(false, a, false, b, (short)0, acc1, false, false);
    b.x = brow0[32]; b.y = brow1[32];
    acc2 = __builtin_amdgcn_wmma_f32_16x16x4_f32(false, a, false, b, (short)0, acc2, false, false);
    b.x = brow0[48]; b.y = brow1[48];
    acc3 = __builtin_amdgcn_wmma_f32_16x16x4_f32(false, a, false, b, (short)0, acc3, false, false);
  }

  // D layout: lane n = tile_n + l16 ; VGPR v holds row m = m0 + v + 8*half
  #pragma unroll
  for (int t = 0; t < 4; ++t) {
    v8f acc = (t==0) ? acc0 : (t==1) ? acc1 : (t==2) ? acc2 : acc3;
    const int n = n0 + t*16 + l16;
    #pragma unroll
    for (int v = 0; v < 8; ++v) {
      const int m = m0 + v + 8*half;
      float val = acc[v];
      if (addMode == 1)      val += rowAdd[m];
      else if (addMode == 2) val += rowAdd[m*4 + (n >> 14)];
      C[(size_t)m * Nn + n] = val;
    }
  }
}

// ================= small-M dense layers =================
__global__ void gemm_smallM(const float* __restrict__ X, const float* __restrict__ W,
                            const float* __restrict__ bias, float* __restrict__ Y,
                            int Mb, int K, int Nn, int relu)
{
  int t = blockIdx.x * blockDim.x + threadIdx.x;
  if (t >= Mb * Nn) return;
  int b = t / Nn, n = t - b * Nn;
  const float* x = X + (size_t)b * K;
  const float* w = W + n;
  float s = bias[n];
  for (int k = 0; k < K; ++k) s = fmaf(x[k], w[(size_t)k * Nn], s);
  if (relu) s = fmaxf(s, 0.f);
  Y[t] = s;
}

__global__ void tanh_half(const float* __restrict__ in, float* __restrict__ o1,
                          float* __restrict__ o2, int n)
{
  int t = blockIdx.x * blockDim.x + threadIdx.x;
  if (t >= n) return;
  float v = tanhf(in[t]) * 0.5f;
  o1[t] = v; o2[t] = v;
}

// ================= pairwise sqdist + iterative top-k-smallest =================
// one workgroup (256 thr) per (b,n) row; d2 row staged in LDS; tie -> lower index (matches top_k)
__global__ __launch_bounds__(256)
void pairwise_topk(const float* __restrict__ pts, int Kk,
                   int* __restrict__ idxOut, float* __restrict__ distOut)
{
  __shared__ float srow[NPTS];
  __shared__ float rv[256];
  __shared__ int   ri[256];
  const int bn = blockIdx.x;
  const int b = bn / NPTS, n = bn % NPTS;
  const float* base = pts + (size_t)b * NPTS * 3;
  const float xn0 = base[n*3], xn1 = base[n*3+1], xn2 = base[n*3+2];
  const float sqn = xn0*xn0 + xn1*xn1 + xn2*xn2;
  for (int m = threadIdx.x; m < NPTS; m += 256) {
    float y0 = base[m*3], y1 = base[m*3+1], y2 = base[m*3+2];
    float d = sqn + (y0*y0+y1*y1+y2*y2) - 2.f*(xn0*y0 + xn1*y1 + xn2*y2);
    d = fmaxf(d, 0.f);
    if (m == n) d = 0.f;
    srow[m] = d;
  }
  __syncthreads();
  for (int j = 0; j < Kk; ++j) {
    float best = BIGF; int bi = NPTS;
    for (int m = threadIdx.x; m < NPTS; m += 256) {
      float v = srow[m];
      if (v < best || (v == best && m < bi)) { best = v; bi = m; }
    }
    rv[threadIdx.x] = best; ri[threadIdx.x] = bi;
    __syncthreads();
    for (int s = 128; s > 0; s >>= 1) {
      if (threadIdx.x < s) {
        float v2 = rv[threadIdx.x+s]; int i2 = ri[threadIdx.x+s];
        if (v2 < rv[threadIdx.x] || (v2 == rv[threadIdx.x] && i2 < ri[threadIdx.x])) {
          rv[threadIdx.x] = v2; ri[threadIdx.x] = i2;
        }
      }
      __syncthreads();
    }
    if (threadIdx.x == 0) {
      int sel = ri[0];
      float d2v = rv[0];
      idxOut[(size_t)bn * Kk + j]  = sel;
      distOut[(size_t)bn * Kk + j] = d2v > 0.f ? sqrtf(d2v) : 0.f;
      srow[sel] = BIGF;
    }
    __syncthreads();
  }
}

// den = dist[1..15]; dense = mean(den)
__global__ void density_finalize(const float* __restrict__ dist16, float* __restrict__ dAll,
                                 float* __restrict__ dMean, float* __restrict__ outMean)
{
  int bn = blockIdx.x * blockDim.x + threadIdx.x;
  if (bn >= PTOT) return;
  const float* d = dist16 + (size_t)bn * 16;
  float s = 0.f;
  for (int c = 0; c < 15; ++c) { float v = d[c+1]; dAll[(size_t)bn*15 + c] = v; s += v; }
  float m = s * (1.f/15.f);
  dMean[bn] = m;
  if (outMean) outMean[bn] = m;
}

// ================= PCA normals (analytic symmetric 3x3 smallest eigvec) ============
__global__ void est_normals(const float* __restrict__ pts, const int* __restrict__ idx32,
                            float* __restrict__ nrm)
{
  int bn = blockIdx.x * blockDim.x + threadIdx.x;
  if (bn >= PTOT) return;
  int b = bn / NPTS, n = bn - b*NPTS;
  const float* base = pts + (size_t)b * NPTS * 3;
  const int* id = idx32 + (size_t)bn * 32;
  float mx = 0, my = 0, mz = 0;
  for (int j = 0; j < 32; ++j) {
    const float* p = base + (size_t)id[j]*3;
    mx += p[0]; my += p[1]; mz += p[2];
  }
  mx *= (1.f/32.f); my *= (1.f/32.f); mz *= (1.f/32.f);
  float a00=0,a01=0,a02=0,a11=0,a12=0,a22=0;
  for (int j = 0; j < 32; ++j) {
    const float* p = base + (size_t)id[j]*3;
    float cx = p[0]-mx, cy = p[1]-my, cz = p[2]-mz;
    a00 += cx*cx; a01 += cx*cy; a02 += cx*cz;
    a11 += cy*cy; a12 += cy*cz; a22 += cz*cz;
  }
  a00*= (1.f/32.f); a01*= (1.f/32.f); a02*= (1.f/32.f);
  a11*= (1.f/32.f); a12*= (1.f/32.f); a22*= (1.f/32.f);
  float vx, vy, vz;
  float p1 = a01*a01 + a02*a02 + a12*a12;
  if (p1 < 1e-30f) {
    vx = 1.f; vy = 0.f; vz = 0.f;
    if (a11 <= a00 && a11 <= a22)      { vx=0.f; vy=1.f; vz=0.f; }
    else if (a22 <= a00 && a22 <= a11) { vx=0.f; vy=0.f; vz=1.f; }
  } else {
    float q  = (a00+a11+a22)*(1.f/3.f);
    float p2 = (a00-q)*(a00-q) + (a11-q)*(a11-q) + (a22-q)*(a22-q) + 2.f*p1;
    float pp = sqrtf(p2*(1.f/6.f));
    float inv = 1.f/pp;
    float b00=(a00-q)*inv, b01=a01*inv, b02=a02*inv;
    float b11=(a11-q)*inv, b12=a12*inv, b22=(a22-q)*inv;
    float detB = b00*(b11*b22-b12*b12) - b01*(b01*b22-b12*b02) + b02*(b01*b12-b11*b02);
    float r = fminf(1.f, fmaxf(-1.f, detB*0.5f));
    float phi = acosf(r)*(1.f/3.f);
    float lam = q + 2.f*pp*cosf(phi + 2.0943951023931953f);   // smallest eigenvalue
    float r0x=a00-lam, r0y=a01,     r0z=a02;
    float r1x=a01,     r1y=a11-lam, r1z=a12;
    float r2x=a02,     r2y=a12,     r2z=a22-lam;
    float c0x=r0y*r1z-r0z*r1y, c0y=r0z*r1x-r0x*r1z, c0z=r0x*r1y-r0y*r1x;
    float c1x=r0y*r2z-r0z*r2y, c1y=r0z*r2x-r0x*r2z, c1z=r0x*r2y-r0y*r2x;
    float c2x=r1y*r2z-r1z*r2y, c2y=r1z*r2x-r1x*r2z, c2z=r1x*r2y-r1y*r2x;
    float n0=c0x*c0x+c0y*c0y+c0z*c0z;
    float n1=c1x*c1x+c1y*c1y+c1z*c1z;
    float n2=c2x*c2x+c2y*c2y+c2z*c2z;
    vx=c0x; vy=c0y; vz=c0z; float bestn=n0;
    if (n1 > bestn) { vx=c1x; vy=c1y; vz=c1z; bestn=n1; }
    if (n2 > bestn) { vx=c2x; vy=c2y; vz=c2z; bestn=n2; }
    if (bestn < 1e-30f) { vx=0.f; vy=0.f; vz=1.f; bestn=1.f; }
    float invn = rsqrtf(bestn);
    vx*=invn; vy*=invn; vz*=invn;
  }
  const float xx = base[n*3], xy = base[n*3+1], xz = base[n*3+2];
  float proj = 0.f;
  for (int j = 0; j < 32; ++j) {
    const float* p = base + (size_t)id[j]*3;
    proj += (p[0]-xx)*vx + (p[1]-xy)*vy + (p[2]-xz)*vz;
  }
  if (proj < 0.f) { vx=-vx; vy=-vy; vz=-vz; }
  nrm[(size_t)bn*3+0]=vx; nrm[(size_t)bn*3+1]=vy; nrm[(size_t)bn*3+2]=vz;
}

// penalty = 1 - cos(n_a, n_j) over 24 NN; std with ddof=1
__global__ void normal_penalty(const float* __restrict__ nrm, const int* __restrict__ idx24,
                               float* __restrict__ penAll, float* __restrict__ stdv,
                               float* __restrict__ outStd)
{
  int bn = blockIdx.x * blockDim.x + threadIdx.x;
  if (bn >= PTOT) return;
  int b = bn / NPTS;
  const float* nb = nrm + (size_t)b * NPTS * 3;
  const int* id = idx24 + (size_t)bn * 24;
  int i0 = id[0];
  float ax = nb[i0*3], ay = nb[i0*3+1], az = nb[i0*3+2];
  float na = fmaxf(sqrtf(ax*ax+ay*ay+az*az), 1e-6f);
  float pen[24];
  float s = 0.f;
  for (int j = 0; j < 24; ++j) {
    int ij = id[j];
    float gx = nb[ij*3], gy = nb[ij*3+1], gz = nb[ij*3+2];
    float ng = fmaxf(sqrtf(gx*gx+gy*gy+gz*gz), 1e-6f);
    float p = 1.f - (ax*gx+ay*gy+az*gz)/(na*ng);
    pen[j] = p;
    penAll[(size_t)bn*24 + j] = p;
    s += p;
  }
  float m = s * (1.f/24.f);
  float v = 0.f;
  for (int j = 0; j < 24; ++j) { float d = pen[j]-m; v += d*d; }
  float sd = sqrtf(v * (1.f/23.f));
  stdv[bn] = sd;
  if (outStd) outStd[bn] = sd;
}

// conv1x1 projection: out[b][o][n] = bias[o] + sum_c W[o][c]*feat[b*N+n][c]
__global__ void conv_proj(const float* __restrict__ feat, int C,
                          const float* __restrict__ W, const float* __restrict__ bias,
                          float* __restrict__ out)
{
  int t = blockIdx.x * blockDim.x + threadIdx.x;
  if (t >= BATCH*128*NPTS) return;
  int n = t % NPTS; int o = (t / NPTS) % 128; int b = t / (128*NPTS);
  const float* f = feat + ((size_t)b*NPTS + n) * C;
  const float* w = W + (size_t)o * C;
  float s = bias[o];
  for (int c = 0; c < C; ++c) s = fmaf(w[c], f[c], s);
  out[t] = s;   // t == (b*128+o)*NPTS + n
}

// score = q.k over 128 ch; softmax is monotonic per row -> top-16 of logits == top-16 of softmax
__global__ __launch_bounds__(256)
void match_topk(const float* __restrict__ q, const float* __restrict__ km,
                const float* __restrict__ vals, float* __restrict__ outV)
{
  __shared__ float qv[128];
  __shared__ float srow[NPTS];
  __shared__ float rv[256];
  __shared__ int   ri[256];
  const int bn = blockIdx.x;
  const int b = bn / NPTS, n = bn % NPTS;
  const float* qb = q + (size_t)b*128*NPTS;
  for (int c = threadIdx.x; c < 128; c += 256) qv[c] = qb[(size_t)c*NPTS + n];
  __syncthreads();
  const float* kb = km + (size_t)b*128*NPTS;
  for (int m = threadIdx.x; m < NPTS; m += 256) {
    float s = 0.f;
    #pragma unroll 8
    for (int c = 0; c < 128; ++c) s = fmaf(qv[c], kb[(size_t)c*NPTS + m], s);
    srow[m] = s;
  }
  __syncthreads();
  for (int j = 0; j < 16; ++j) {
    float best = -BIGF; int bi = NPTS;
    for (int m = threadIdx.x; m < NPTS; m += 256) {
      float v = srow[m];
      if (v > best || (v == best && m < bi)) { best = v; bi = m; }
    }
    rv[threadIdx.x] = best; ri[threadIdx.x] = bi;
    __syncthreads();
    for (int s2 = 128; s2 > 0; s2 >>= 1) {
      if (threadIdx.x < s2) {
        float v2 = rv[threadIdx.x+s2]; int i2 = ri[threadIdx.x+s2];
        if (v2 > rv[threadIdx.x] || (v2 == rv[threadIdx.x] && i2 < ri[threadIdx.x])) {
          rv[threadIdx.x] = v2; ri[threadIdx.x] = i2;
        }
      }
      __syncthreads();
    }
    if (threadIdx.x == 0) {
      int sel = ri[0];
      outV[(size_t)bn*16 + j] = vals[(size_t)b*NPTS + sel];
      srow[sel] = -BIGF;
    }
    __syncthreads();
  }
}

// midpoints to the 8 NN (excl. self) of coarse_van -> point_feat [B][3][PNUM]
__global__ void upsample_mid(const float* __restrict__ cvan, const int* __restrict__ idx9,
                             float* __restrict__ pf)
{
  int t = blockIdx.x * blockDim.x + threadIdx.x;
  if (t >= PTOT*8) return;
  int r = t & 7; int bn = t >> 3;
  int b = bn / NPTS, n = bn % NPTS;
  const float* base = cvan + (size_t)b*NPTS*3;
  int id = idx9[(size_t)bn*9 + 1 + r];
  int i = n*8 + r;
  for (int c = 0; c < 3; ++c) {
    float x = base[n*3+c];
    float nb = base[id*3+c];
    pf[((size_t)b*3 + c)*PNUM + i] = x + (nb - x)*0.5f;
  }
}

// compact per-point feature matrix Fc[KPAD][PALL]  (channels 0..36 real, 37..39 zero)
__global__ void build_feat(const float* __restrict__ dtopk, const float* __restrict__ ntopk,
                           const float* __restrict__ denseC, const float* __restrict__ stdC,
                           const float* __restrict__ pf, float* __restrict__ Fc)
{
  int p = blockIdx.x * blockDim.x + threadIdx.x;
  if (p >= PALL) return;
  int b = p >> 14; int i = p & (PNUM-1); int n = i >> 3;
  size_t bn = (size_t)b*NPTS + n;
  for (int c = 0; c < 16; ++c) Fc[(size_t)c*PALL + p]      = dtopk[bn*16 + c];  // repeat(.,8)
  for (int c = 0; c < 16; ++c) Fc[(size_t)(16+c)*PALL + p] = ntopk[bn*16 + c];
  int imod = i & (NPTS-1);                                   // jnp.tile(...,(1,8)) -> i % 2048
  Fc[(size_t)32*PALL + p] = denseC[(size_t)b*NPTS + imod];
  Fc[(size_t)33*PALL + p] = stdC[(size_t)b*NPTS + imod];
  for (int c = 0; c < 3; ++c) Fc[(size_t)(34+c)*PALL + p] = pf[((size_t)b*3 + c)*PNUM + i];
  for (int c = 37; c < KPAD; ++c) Fc[(size_t)c*PALL + p] = 0.f;
}

// compact fc1 weight: cols {0..31, 1056..1060} of fc1_w, zero-padded to KPAD
__global__ void build_wc(const float* __restrict__ fc1w, float* __restrict__ Wc)
{
  int t = blockIdx.x * blockDim.x + threadIdx.x;
  if (t >= C1*KPAD) return;
  int o = t / KPAD, c = t % KPAD;
  float v = 0.f;
  if (c < 32)      v = fc1w[(size_t)o*1061 + c];
  else if (c < 37) v = fc1w[(size_t)o*1061 + 1024 + c];     // 32->1056 .. 36->1060
  Wc[t] = v;
}

// g[o][b] = fc1_b[o] + fc1_w[o, 32:1056] . z[b]   (broadcast latent term)
__global__ void build_g(const float* __restrict__ fc1w, const float* __restrict__ fc1b,
                        const float* __restrict__ z, float* __restrict__ g)
{
  int t = blockIdx.x * blockDim.x + threadIdx.x;
  if (t >= C1*BATCH) return;
  int o = t / BATCH, b = t % BATCH;
  const float* wz = fc1w + (size_t)o*1061 + 32;
  const float* zz = z + (size_t)b*1024;
  float s = fc1b[o];
  for (int c = 0; c < 1024; ++c) s = fmaf(wz[c], zz[c], s);
  g[t] = s;
}

// per-channel batchnorm stats over PALL samples (population var)
__global__ __launch_bounds__(256)
void bn_stats(const float* __restrict__ Y, float* __restrict__ mean, float* __restrict__ var)
{
  __shared__ float ss[256], sq[256];
  int ch = blockIdx.x;
  const float* y = Y + (size_t)ch * PALL;
  float s = 0.f, q2 = 0.f;
  for (int p = threadIdx.x; p < PALL; p += 256) { float v = y[p]; s += v; q2 += v*v; }
  ss[threadIdx.x] = s; sq[threadIdx.x] = q2;
  __syncthreads();
  for (int st = 128; st > 0; st >>= 1) {
    if (threadIdx.x < st) { ss[threadIdx.x] += ss[threadIdx.x+st]; sq[threadIdx.x] += sq[threadIdx.x+st]; }
    __syncthreads();
  }
  if (threadIdx.x == 0) {
    float m = ss[0] * (1.f/(float)PALL);
    mean[ch] = m;
    var[ch]  = fmaxf(sq[0] * (1.f/(float)PALL) - m*m, 0.f);
  }
}

__global__ void bn_relu(float* __restrict__ Y, const float* __restrict__ mean,
                        const float* __restrict__ var, const float* __restrict__ gam,
                        const float* __restrict__ bet)
{
  size_t t = (size_t)blockIdx.x * blockDim.x + threadIdx.x;
  if (t >= (size_t)C1 * PALL) return;
  int ch = (int)(t / PALL);
  float v = (Y[t] - mean[ch]) * rsqrtf(var[ch] + 1e-5f) * gam[ch] + bet[ch];
  Y[t] = fmaxf(v, 0.f);
}

// fc3 (3x512) + fine = tanh(point_feat + offset)*0.5, written as (b, 16384, 3)
__global__ void fc3_fine(const float* __restrict__ x2, const float* __restrict__ w3,
                         const float* __restrict__ b3, const float* __restrict__ pf,
                         float* __restrict__ fineOut)
{
  int p = blockIdx.x * blockDim.x + threadIdx.x;
  if (p >= PALL) return;
  float s0 = b3[0], s1 = b3[1], s2 = b3[2];
  for (int k = 0; k < C1; ++k) {
    float v = x2[(size_t)k*PALL + p];
    s0 = fmaf(w3[k],        v, s0);
    s1 = fmaf(w3[512 + k],  v, s1);
    s2 = fmaf(w3[1024 + k], v, s2);
  }
  int b = p >> 14; int i = p & (PNUM-1);
  size_t ob = (size_t)b*PNUM*3 + (size_t)i*3;
  fineOut[ob + 0] = tanhf(pf[((size_t)b*3+0)*PNUM + i] + s0) * 0.5f;
  fineOut[ob + 1] = tanhf(pf[((size_t)b*3+1)*PNUM + i] + s1) * 0.5f;
  fineOut[ob + 2] = tanhf(pf[((size_t)b*3+2)*PNUM + i] + s2) * 0.5f;
}

// ================================ host ================================
static inline size_t alignup(size_t x) { return (x + 255) & ~(size_t)255; }

extern "C" void kernel_launch(void* const* d_in, const int* in_sizes, int n_in,
                              void* d_out, int out_size, void* d_ws, size_t ws_size,
                              hipStream_t stream) {
  (void)in_sizes; (void)n_in; (void)out_size; (void)ws_size;
  const float* z     = (const float*)d_in[0];
  const float* input = (const float*)d_in[1];
  const float* w1 = (const float*)d_in[2];  const float* b1 = (const float*)d_in[3];
  const float* w2 = (const float*)d_in[4];  const float* b2 = (const float*)d_in[5];
  const float* w3 = (const float*)d_in[6];  const float* b3 = (const float*)d_in[7];
  const float* w4 = (const float*)d_in[8];  const float* b4 = (const float*)d_in[9];
  const float* pdc_w = (const float*)d_in[10]; const float* pdc_b = (const float*)d_in[11];
  const float* pdi_w = (const float*)d_in[12]; const float* pdi_b = (const float*)d_in[13];
  const float* pnc_w = (const float*)d_in[14]; const float* pnc_b = (const float*)d_in[15];
  const float* pni_w = (const float*)d_in[16]; const float* pni_b = (const float*)d_in[17];
  const float* fc1_w = (const float*)d_in[18]; const float* fc1_b = (const float*)d_in[19];
  const float* bn1_g = (const float*)d_in[20]; const float* bn1_b = (const float*)d_in[21];
  const float* fc2_w = (const float*)d_in[22]; const float* fc2_b = (const float*)d_in[23];
  const float* bn2_g = (const float*)d_in[24]; const float* bn2_b = (const float*)d_in[25];
  const float* fc3_w = (const float*)d_in[26]; const float* fc3_b = (const float*)d_in[27];
  float* out = (float*)d_out;

  // ---- workspace bump allocator ----
  char* ws = (char*)d_ws; size_t off = 0;
  auto allocf = [&](size_t nfloat) { size_t o = off; off = alignup(off + nfloat*sizeof(float)); return (float*)(ws + o); };
  auto alloci = [&](size_t nint)   { size_t o = off; off = alignup(off + nint*sizeof(int));   return (int*)(ws + o); };

  float* h1   = allocf((size_t)BATCH*2048);
  float* h2   = allocf((size_t)BATCH*2048);
  float* h3   = allocf((size_t)BATCH*2048);
  float* cvan = allocf((size_t)BATCH*6144);
  float* crs  = allocf((size_t)BATCH*6144);
  int*   idx16_c = alloci((size_t)PTOT*16); float* d16_c = allocf((size_t)PTOT*16);
  int*   idx32_c = alloci((size_t)PTOT*32); float* d32_c = allocf((size_t)PTOT*32);
  int*   idx24_c = alloci((size_t)PTOT*24); float* d24_c = allocf((size_t)PTOT*24);
  int*   idx16_i = alloci((size_t)PTOT*16); float* d16_i = allocf((size_t)PTOT*16);
  int*   idx32_i = alloci((size_t)PTOT*32); float* d32_i = allocf((size_t)PTOT*32);
  int*   idx24_i = alloci((size_t)PTOT*24); float* d24_i = allocf((size_t)PTOT*24);
  int*   idx9_v  = alloci((size_t)PTOT*9);  float* d9_v  = allocf((size_t)PTOT*9);
  float* dAll_c = allocf((size_t)PTOT*15);  float* dense_c = allocf(PTOT);
  float* dAll_i = allocf((size_t)PTOT*15);  float* dense_i = allocf(PTOT);
  float* nrm_c  = allocf((size_t)PTOT*3);   float* nrm_i   = allocf((size_t)PTOT*3);
  float* pen_c  = allocf((size_t)PTOT*24);  float* std_c   = allocf(PTOT);
  float* pen_i  = allocf((size_t)PTOT*24);  float* std_i   = allocf(PTOT);
  float* dap = allocf((size_t)BATCH*128*NPTS);
  float* dip = allocf((size_t)BATCH*128*NPTS);
  float* nap = allocf((size_t)BATCH*128*NPTS);
  float* nip = allocf((size_t)BATCH*128*NPTS);
  float* dtopk = allocf((size_t)PTOT*16);
  float* ntopk = allocf((size_t)PTOT*16);
  float* pf  = allocf((size_t)BATCH*3*PNUM);
  float* Fc  = allocf((size_t)KPAD*PALL);
  float* Wc  = allocf((size_t)C1*KPAD);
  float* gz  = allocf((size_t)C1*BATCH);
  float* y1  = allocf((size_t)C1*PALL);
  float* y2  = allocf((size_t)C1*PALL);
  float* bnm = allocf(C1); float* bnv = allocf(C1);

  // ---- 1) coarse MLP ----
  gemm_smallM<<<(BATCH*2048+255)/256, 256, 0, stream>>>(z,  w1, b1, h1, BATCH, 1024, 2048, 1);
  gemm_smallM<<<(BATCH*2048+255)/256, 256, 0, stream>>>(h1, w2, b2, h2, BATCH, 2048, 2048, 1);
  gemm_smallM<<<(BATCH*2048+255)/256, 256, 0, stream>>>(h2, w3, b3, h3, BATCH, 2048, 2048, 1);
  gemm_smallM<<<(BATCH*6144+255)/256, 256, 0, stream>>>(h3, w4, b4, cvan, BATCH, 2048, 6144, 0);
  tanh_half<<<(BATCH*6144+255)/256, 256, 0, stream>>>(cvan, crs, out + OUT_COARSE, BATCH*6144);

  // ---- 2) neighborhoods ----
  pairwise_topk<<<PTOT, 256, 0, stream>>>(crs,   16, idx16_c, d16_c);
  pairwise_topk<<<PTOT, 256, 0, stream>>>(crs,   32, idx32_c, d32_c);
  pairwise_topk<<<PTOT, 256, 0, stream>>>(crs,   24, idx24_c, d24_c);
  pairwise_topk<<<PTOT, 256, 0, stream>>>(input, 16, idx16_i, d16_i);
  pairwise_topk<<<PTOT, 256, 0, stream>>>(input, 32, idx32_i, d32_i);
  pairwise_topk<<<PTOT, 256, 0, stream>>>(input, 24, idx24_i, d24_i);
  pairwise_topk<<<PTOT, 256, 0, stream>>>(cvan,   9, idx9_v,  d9_v);

  // ---- 3) density / normals ----
  density_finalize<<<(PTOT+255)/256, 256, 0, stream>>>(d16_c, dAll_c, dense_c, out + OUT_DENSE);
  density_finalize<<<(PTOT+255)/256, 256, 0, stream>>>(d16_i, dAll_i, dense_i, (float*)nullptr);
  est_normals<<<(PTOT+255)/256, 256, 0, stream>>>(crs,   idx32_c, nrm_c);
  est_normals<<<(PTOT+255)/256, 256, 0, stream>>>(input, idx32_i, nrm_i);
  normal_penalty<<<(PTOT+255)/256, 256, 0, stream>>>(nrm_c, idx24_c, pen_c, std_c, out + OUT_NORM);
  normal_penalty<<<(PTOT+255)/256, 256, 0, stream>>>(nrm_i, idx24_i, pen_i, std_i, (float*)nullptr);

  // ---- 4) projections + matching (softmax skipped: monotonic per row) ----
  const int projT = BATCH*128*NPTS;
  conv_proj<<<(projT+255)/256, 256, 0, stream>>>(dAll_c, 15, pdc_w, pdc_b, dap);
  conv_proj<<<(projT+255)/256, 256, 0, stream>>>(dAll_i, 15, pdi_w, pdi_b, dip);
  conv_proj<<<(projT+255)/256, 256, 0, stream>>>(pen_c,  24, pnc_w, pnc_b, nap);
  conv_proj<<<(projT+255)/256, 256, 0, stream>>>(pen_i,  24, pni_w, pni_b, nip);
  match_topk<<<PTOT, 256, 0, stream>>>(dap, dip, dense_i, dtopk);
  match_topk<<<PTOT, 256, 0, stream>>>(nap, nip, std_i,   ntopk);

  // ---- 5) upsample + feature assembly ----
  upsample_mid<<<(PTOT*8+255)/256, 256, 0, stream>>>(cvan, idx9_v, pf);
  build_feat<<<(PALL+255)/256, 256, 0, stream>>>(dtopk, ntopk, dense_c, std_c, pf, Fc);
  build_wc<<<(C1*KPAD+255)/256, 256, 0, stream>>>(fc1_w, Wc);
  build_g<<<(C1*BATCH+255)/256, 256, 0, stream>>>(fc1_w, fc1_b, z, gz);

  // ---- 6) fc1/fc2 via fp32 WMMA, batchnorm, relu ----
  dim3 gg(PALL/64, C1/16), bb(32);
  wmma_gemm_f32<<<gg, bb, 0, stream>>>(Wc, Fc, y1, gz, C1, KPAD, PALL, 2);
  bn_stats<<<C1, 256, 0, stream>>>(y1, bnm, bnv);
  bn_relu<<<(unsigned)(((size_t)C1*PALL + 255)/256), 256, 0, stream>>>(y1, bnm, bnv, bn1_g, bn1_b);
  wmma_gemm_f32<<<gg, bb, 0, stream>>>(fc2_w, y1, y2, fc2_b, C1, C1, PALL, 1);
  bn_stats<<<C1, 256, 0, stream>>>(y2, bnm, bnv);
  bn_relu<<<(unsigned)(((size_t)C1*PALL + 255)/256), 256, 0, stream>>>(y2, bnm, bnv, bn2_g, bn2_b);

  // ---- 7) fc3 + fine ----
  fc3_fine<<<(PALL+255)/256, 256, 0, stream>>>(y2, fc3_w, fc3_b, pf, out + OUT_FINE);
}